// ORec_89026082111512
// MI455X (gfx1250) — compile-verified
//
#include <hip/hip_runtime.h>
#include <math.h>

#define BATCH 4096
#define SEQL  200
#define DIM   128
#define NEGC  (-4294967296.0f)   // -2^32
#define EPS_A 0.999f
#define EPS_B 0.0001f

typedef __attribute__((ext_vector_type(2))) float v2f;
typedef __attribute__((ext_vector_type(8))) float v8f;

// ---------------------------------------------------------------------------
// Tile GEMM, all dims K=N=128, M multiple of 16.
//   WT=true : D[m,n] = sum_k A[m,k] * W[n,k]  (+bias[n])     (A @ W^T)
//   WT=false: D[m,n] = sum_k A[m,k] * W[k,n]  (+bias[n])     (A @ W)
//   GATHER  : A row m is emb_table[gidx[m]] instead of A[m].
// One wave per 16x16 output tile; block = 8 waves covers all 8 N-tiles.
// f32 WMMA layouts (ISA 7.12.2): A 16x4: lane r=L%16 holds row r, VGPR pair =
// K = {2h, 2h+1} with h=L/16.  B 4x16 mirrors it.  C/D: acc[i] = D[i+8h, r].
// ---------------------------------------------------------------------------
template <bool WT, bool GATHER>
__global__ __launch_bounds__(256) void gemm128_wmma(
    const float* __restrict__ A, const float* __restrict__ W,
    const float* __restrict__ bias, const int* __restrict__ gidx,
    const float* __restrict__ table, float* __restrict__ D) {
  const int lane = threadIdx.x & 31;
  const int wave = threadIdx.x >> 5;
  const int r = lane & 15;       // row index within tile (A rows / W rows)
  const int h = lane >> 4;       // K half
  const int m0 = blockIdx.x * 16;
  const int n0 = wave * 16;

  const float* arow;
  if (GATHER) {
    arow = table + (size_t)gidx[m0 + r] * DIM;
  } else {
    arow = A + (size_t)(m0 + r) * DIM;
  }
  const float* wrow = W + (size_t)(n0 + r) * DIM;  // used when WT

  v8f acc = {};
#pragma unroll
  for (int kk = 0; kk < DIM; kk += 4) {
    v2f a = *(const v2f*)(arow + kk + 2 * h);
    v2f b;
    if (WT) {
      b = *(const v2f*)(wrow + kk + 2 * h);
    } else {
      b.x = W[(size_t)(kk + 2 * h) * DIM + n0 + r];
      b.y = W[(size_t)(kk + 2 * h + 1) * DIM + n0 + r];
    }
    acc = __builtin_amdgcn_wmma_f32_16x16x4_f32(
        /*neg_a=*/false, a, /*neg_b=*/false, b,
        /*c_mod=*/(short)0, acc, /*reuse_a=*/false, /*reuse_b=*/false);
  }

  const float bv = bias ? bias[n0 + r] : 0.0f;
#pragma unroll
  for (int i = 0; i < 8; ++i) {
    D[(size_t)(m0 + i + 8 * h) * DIM + n0 + r] = acc[i] + bv;
  }
}

// c_b = q_b . bk  (the additive logit constant; mask MULTIPLIES by -2^32, so
// it must be kept exactly)
__global__ __launch_bounds__(256) void cq_kernel(const float* __restrict__ q,
                                                 const float* __restrict__ bk,
                                                 float* __restrict__ cq) {
  int b = blockIdx.x * blockDim.x + threadIdx.x;
  if (b >= BATCH) return;
  float acc = 0.0f;
  const float* row = q + (size_t)b * DIM;
#pragma unroll 8
  for (int i = 0; i < DIM; ++i) acc += row[i] * bk[i];
  cq[b] = acc;
}

// One workgroup per batch element: logits -> mask -> softmax -> weighted
// embedding sum (eagg). Embedding table is L2-resident (51MB < 192MB L2).
__global__ __launch_bounds__(256) void attn_kernel(
    const int* __restrict__ hist, const float* __restrict__ emb,
    const float* __restrict__ qk, const float* __restrict__ cq,
    float* __restrict__ eagg) {
  __shared__ float s_qk[DIM];
  __shared__ float s_red[256];
  __shared__ float s_score[SEQL];
  __shared__ int s_idx[SEQL];
  __shared__ float s_part[DIM];

  const int tid = threadIdx.x;
  const int b = blockIdx.x;

  if (tid < DIM) s_qk[tid] = qk[(size_t)b * DIM + tid];
  const float c = cq[b];

  const bool active = tid < SEQL;
  int idx = 0;
  if (active) idx = hist[(size_t)b * SEQL + tid];
  if (active) s_idx[tid] = idx;
  __syncthreads();

  // ---- pass 1: logits[l] = qk . e_l + c, multiplicative mask -------------
  float lg = -INFINITY;
  if (active) {
    const float* e = emb + (size_t)idx * DIM;
    __builtin_prefetch(e, 0, 1);  // global_prefetch_b8
    float acc = 0.0f;
    const float4* e4 = (const float4*)e;
    const float4* q4 = (const float4*)s_qk;
#pragma unroll
    for (int d = 0; d < DIM / 4; ++d) {
      float4 ev = e4[d];
      float4 qv = q4[d];
      acc += ev.x * qv.x + ev.y * qv.y + ev.z * qv.z + ev.w * qv.w;
    }
    lg = acc + c;
    if (idx == 0) lg = lg * NEGC;  // reference: logits*(1-mask)*NEG
  }

  // ---- block softmax over 200 logits -------------------------------------
  s_red[tid] = lg;
  __syncthreads();
  for (int s = 128; s > 0; s >>= 1) {
    if (tid < s) s_red[tid] = fmaxf(s_red[tid], s_red[tid + s]);
    __syncthreads();
  }
  const float mx = s_red[0];
  __syncthreads();

  float ex = active ? expf(lg - mx) : 0.0f;
  s_red[tid] = ex;
  __syncthreads();
  for (int s = 128; s > 0; s >>= 1) {
    if (tid < s) s_red[tid] += s_red[tid + s];
    __syncthreads();
  }
  const float inv = 1.0f / s_red[0];
  if (active) s_score[tid] = ex * inv;
  __syncthreads();

  // ---- pass 2: eagg[d] = sum_l score[l] * e_l[d] --------------------------
  const int d = tid & (DIM - 1);
  const int half = tid >> 7;  // 0 handles l in [0,100), 1 handles [100,200)
  float acc = 0.0f;
  const int l0 = half * (SEQL / 2);
  for (int l = l0; l < l0 + SEQL / 2; ++l) {
    acc += s_score[l] * emb[(size_t)s_idx[l] * DIM + d];
  }
  if (half == 1) s_part[d] = acc;
  __syncthreads();
  if (half == 0) eagg[(size_t)b * DIM + d] = acc + s_part[d];
}

// Per-batch loss: z = p @ Wc^T + bc, plus prior logits, log-softmax(2), NLL.
__global__ __launch_bounds__(256) void loss_kernel(
    const float* __restrict__ p, const float* __restrict__ Wc,
    const float* __restrict__ bc, const float* __restrict__ prior,
    const int* __restrict__ label, float* __restrict__ lb) {
  int b = blockIdx.x * blockDim.x + threadIdx.x;
  if (b >= BATCH) return;
  const float* row = p + (size_t)b * DIM;
  float z0 = 0.0f, z1 = 0.0f;
#pragma unroll 4
  for (int i = 0; i < DIM; ++i) {
    float pv = row[i];
    z0 += pv * Wc[i];
    z1 += pv * Wc[DIM + i];
  }
  z0 += bc[0];
  z1 += bc[1];
  const float sc = prior[b];
  const float s0 = (1.0f - sc) * EPS_A + EPS_B;
  const float s1 = sc * EPS_A + EPS_B;
  const float f0 = z0 - logf(1.0f / s0 - 1.0f);
  const float f1 = z1 - logf(1.0f / s1 - 1.0f);
  const float m = fmaxf(f0, f1);
  const float logZ = m + logf(expf(f0 - m) + expf(f1 - m));
  const float fl = (label[b] == 0) ? f0 : f1;
  lb[b] = -(fl - logZ);
}

__global__ __launch_bounds__(1024) void reduce_kernel(
    const float* __restrict__ lb, float* __restrict__ out) {
  __shared__ float s[1024];
  int tid = threadIdx.x;
  float acc = lb[tid] + lb[tid + 1024] + lb[tid + 2048] + lb[tid + 3072];
  s[tid] = acc;
  __syncthreads();
  for (int st = 512; st > 0; st >>= 1) {
    if (tid < st) s[tid] += s[tid + st];
    __syncthreads();
  }
  if (tid == 0) out[0] = s[0] / (float)BATCH;
}

extern "C" void kernel_launch(void* const* d_in, const int* in_sizes, int n_in,
                              void* d_out, int out_size, void* d_ws,
                              size_t ws_size, hipStream_t stream) {
  const int* hist = (const int*)d_in[0];
  const int* cand = (const int*)d_in[1];
  const float* prior = (const float*)d_in[2];
  const int* label = (const int*)d_in[3];
  const float* emb = (const float*)d_in[4];
  const float* Wq = (const float*)d_in[5];
  const float* bq = (const float*)d_in[6];
  const float* Wk = (const float*)d_in[7];
  const float* bk = (const float*)d_in[8];
  const float* Wv = (const float*)d_in[9];
  const float* bv = (const float*)d_in[10];
  const float* Wp = (const float*)d_in[11];
  const float* bp = (const float*)d_in[12];
  const float* Wc = (const float*)d_in[13];
  const float* bc = (const float*)d_in[14];
  float* out = (float*)d_out;

  float* ws = (float*)d_ws;
  float* q_d    = ws;                       // B*DIM
  float* qk_d   = q_d + BATCH * DIM;        // B*DIM
  float* eagg_d = qk_d + BATCH * DIM;       // B*DIM
  float* agg_d  = eagg_d + BATCH * DIM;     // B*DIM
  float* p_d    = agg_d + BATCH * DIM;      // B*DIM
  float* cq_d   = p_d + BATCH * DIM;        // B
  float* lb_d   = cq_d + BATCH;             // B

  // q = gather(emb, cand) @ Wq^T + bq          (WMMA f32)
  gemm128_wmma<true, true><<<BATCH / 16, 256, 0, stream>>>(
      nullptr, Wq, bq, cand, emb, q_d);
  // qk = q @ Wk                                (WMMA f32)
  gemm128_wmma<false, false><<<BATCH / 16, 256, 0, stream>>>(
      q_d, Wk, nullptr, nullptr, nullptr, qk_d);
  // c_b = q_b . bk
  cq_kernel<<<BATCH / 256, 256, 0, stream>>>(q_d, bk, cq_d);
  // attention core: logits -> softmax -> weighted embedding sum
  attn_kernel<<<BATCH, 256, 0, stream>>>(hist, emb, qk_d, cq_d, eagg_d);
  // agg = eagg @ Wv^T + bv                     (WMMA f32)
  gemm128_wmma<true, false><<<BATCH / 16, 256, 0, stream>>>(
      eagg_d, Wv, bv, nullptr, nullptr, agg_d);
  // p = agg @ Wp^T + bp                        (WMMA f32)
  gemm128_wmma<true, false><<<BATCH / 16, 256, 0, stream>>>(
      agg_d, Wp, bp, nullptr, nullptr, p_d);
  // per-sample loss, then deterministic tree reduction
  loss_kernel<<<BATCH / 256, 256, 0, stream>>>(p_d, Wc, bc, prior, label, lb_d);
  reduce_kernel<<<1, 1024, 0, stream>>>(lb_d, out);
}